// AdaptiveLinearWithChannel_81741817577961
// MI455X (gfx1250) — compile-verified
//
#include <hip/hip_runtime.h>
#include <stdint.h>

typedef __attribute__((ext_vector_type(2))) float v2f;
typedef __attribute__((ext_vector_type(8))) float v8f;

#define NCHAN   64
#define NPTS    2048
#define DIN     256
#define DOUT    256
#define NFRAMES 32

#define ROWS    128            // output rows per block (8 waves x 16)
#define KT      16             // K panel depth (double buffered)
#define NPANEL  (DIN / KT)     // 16 panels
#define WSTRIDE 264            // W panel row stride (floats): 2*264 % 64 == 16 -> half-waves on disjoint banks
#define XSTRIDE 20             // x panel row stride (floats): row*20 mod 64 all-distinct bank pairs

// gfx1250 async direct-to-LDS copy: 16B per lane, memory -> LDS, tracked by ASYNCcnt.
// VDST operand carries the LDS byte offset (generic LDS addr low 32 bits == LDS offset).
__device__ __forceinline__ void async_copy16(const void* gaddr, uint32_t lds_off) {
    asm volatile("global_load_async_to_lds_b128 %0, %1, off"
                 :: "v"(lds_off), "v"(gaddr) : "memory");
}
__device__ __forceinline__ void wait_async0() {
    asm volatile("s_wait_asynccnt 0" ::: "memory");
}

__global__ __launch_bounds__(256)
void alwc_wmma_kernel(const float* __restrict__ x,
                      const float* __restrict__ t,
                      const int*   __restrict__ midx,
                      const float* __restrict__ W,
                      const float* __restrict__ B,
                      float*       __restrict__ out)
{
    __shared__ float lw[2][KT * WSTRIDE];    // 2 x 16,896 B
    __shared__ float lx[2][ROWS * XSTRIDE];  // 2 x 10,240 B   (total 54,272 B)

    const int c       = blockIdx.x;
    const int rowBase = blockIdx.y * ROWS;
    const int tid     = threadIdx.x;
    const int lane    = tid & 31;
    const int wave    = tid >> 5;
    const int half    = lane >> 4;     // 0: lanes 0-15, 1: lanes 16-31
    const int l16     = lane & 15;

    const int    wsel = midx[c];
    const float* Wg   = W + (size_t)wsel * DIN * DOUT;
    const float* xc   = x + ((size_t)c * NPTS + rowBase) * DIN;

    int bias_idx = (int)(t[c] * (float)(NFRAMES - 1));   // truncation matches .astype(int32)
    if (bias_idx < 0) bias_idx = 0;
    if (bias_idx > NFRAMES - 1) bias_idx = NFRAMES - 1;
    const float* bg = B + (size_t)bias_idx * DOUT;

    // Stage one K-panel (W: KT x 256, x: 128 x KT) into LDS buffer `buf` via async copies.
    auto stage = [&](int buf, int kp) {
        // W panel: 16 rows x 64 16B-chunks = 1024 chunks, 4 per thread (coalesced rows)
#pragma unroll
        for (int i = 0; i < 4; ++i) {
            int linear = tid + i * 256;
            int row    = linear >> 6;          // 64 chunks per row
            int col4   = linear & 63;
            const float* g = Wg + (size_t)(kp * KT + row) * DOUT + col4 * 4;
            async_copy16(g, (uint32_t)(uintptr_t)&lw[buf][row * WSTRIDE + col4 * 4]);
        }
        // x panel: 128 rows x 4 16B-chunks = 512 chunks, 2 per thread
#pragma unroll
        for (int i = 0; i < 2; ++i) {
            int linear = tid + i * 256;
            int row    = linear >> 2;          // 4 chunks per row
            int col4   = linear & 3;
            const float* g = xc + (size_t)row * DIN + kp * KT + col4 * 4;
            async_copy16(g, (uint32_t)(uintptr_t)&lx[buf][row * XSTRIDE + col4 * 4]);
        }
    };

    v8f acc[16];
#pragma unroll
    for (int i = 0; i < 16; ++i) acc[i] = (v8f)0.0f;

    stage(0, 0);   // prime the pipeline

    for (int kp = 0; kp < NPANEL; ++kp) {
        const int cur = kp & 1;

        wait_async0();        // our own async copies into buffer `cur` are done
        __syncthreads();      // everyone's copies done; everyone done reading buffer cur^1

        if (kp + 1 < NPANEL)  // overlap next panel's DMA with this panel's WMMAs
            stage(cur ^ 1, kp + 1);

#pragma unroll
        for (int k = 0; k < KT; k += 4) {
            const int r0 = k + 2 * half;
            // A fragment (16x4 f32): lane half selects K pair {k,k+1} / {k+2,k+3}
            v2f a = *(const v2f*)(&lx[cur][(wave * 16 + l16) * XSTRIDE + r0]);
#pragma unroll
            for (int nt = 0; nt < 16; ++nt) {
                const int col = nt * 16 + l16;
                v2f b;
                b.x = lw[cur][r0 * WSTRIDE + col];
                b.y = lw[cur][(r0 + 1) * WSTRIDE + col];
                acc[nt] = __builtin_amdgcn_wmma_f32_16x16x4_f32(
                    /*neg_a=*/false, a, /*neg_b=*/false, b,
                    /*c_mod=*/(short)0, acc[nt],
                    /*reuse_a=*/false, /*reuse_b=*/false);
            }
        }
    }

    // ---- epilogue: D layout -> global, fused bias add ----
    float* oc = out + ((size_t)c * NPTS + rowBase + wave * 16) * DOUT;
#pragma unroll
    for (int nt = 0; nt < 16; ++nt) {
        const int col  = nt * 16 + l16;
        const float bb = bg[col];
#pragma unroll
        for (int v = 0; v < 8; ++v) {
            const int row = v + 8 * half;       // VGPR v: M=v (lanes 0-15), M=v+8 (lanes 16-31)
            oc[(size_t)row * DOUT + col] = acc[nt][v] + bb;
        }
    }
}

extern "C" void kernel_launch(void* const* d_in, const int* in_sizes, int n_in,
                              void* d_out, int out_size, void* d_ws, size_t ws_size,
                              hipStream_t stream) {
    const float* x    = (const float*)d_in[0];
    const float* t    = (const float*)d_in[1];
    const int*   midx = (const int*)d_in[2];
    const float* W    = (const float*)d_in[3];
    const float* B    = (const float*)d_in[4];
    float*       out  = (float*)d_out;

    dim3 grid(NCHAN, NPTS / ROWS);   // 64 x 16 blocks
    dim3 block(256);
    alwc_wmma_kernel<<<grid, block, 0, stream>>>(x, t, midx, W, B, out);
}